// TaskAlignedAssigner_32358283608499
// MI455X (gfx1250) — compile-verified
//
#include <hip/hip_runtime.h>
#include <stdint.h>

#define BS      32
#define NA      8400
#define NM      100
#define NC      80
#define TOPK    13
#define EPSF    1e-9f

typedef float v2f __attribute__((ext_vector_type(2)));
typedef float v8f __attribute__((ext_vector_type(8)));

// ---------------------------------------------------------------- init
__global__ void k_init(int* pos_sum, int* agt, unsigned* pos_align, unsigned* pos_ovl) {
    int i = blockIdx.x * blockDim.x + threadIdx.x;
    if (i < BS * NA) {
        pos_sum[i] = 0;
        agt[i] = 0x7fffffff;
    }
    if (i < BS * NM) {
        pos_align[i] = 0u;
        pos_ovl[i] = 0u;
    }
}

// ---------------------------------------------------------------- tile pass
// One wave (32 threads) per 16(gt) x 16(anchor) tile of one batch.
// bbox_scores tile = OneHot(labels[16 x 80]) @ pred_scores[16 x 80]^T via
// 20 x v_wmma_f32_16x16x4_f32 (exact in fp32), then IoU + in_gts + metric.
__global__ void __launch_bounds__(32)
k_tile(const float* __restrict__ pred_scores,
       const float* __restrict__ pred_bboxes,
       const float* __restrict__ anchor_points,
       const int*   __restrict__ gt_labels,
       const float* __restrict__ gt_bboxes,
       float* __restrict__ overlaps,
       float* __restrict__ metrics) {
    const int b   = blockIdx.z;
    const int g0  = blockIdx.y * 16;
    const int a0  = blockIdx.x * 16;
    const int lane = threadIdx.x;
    const int mrow = lane & 15;
    const bool hi  = lane >= 16;

    __shared__ float sGt[16 * 4];
    __shared__ int   sLab[16];

    // Stage GT boxes (256B) + labels (64B) into LDS via CDNA5 async copy.
    if (lane < 16) {
        int g = g0 + lane; if (g > NM - 1) g = NM - 1;
        const void* src = (const void*)(gt_bboxes + ((size_t)b * NM + g) * 4);
        uint32_t lo = (uint32_t)(uintptr_t)(&sGt[lane * 4]);
        asm volatile("global_load_async_to_lds_b128 %0, %1, off"
                     :: "v"(lo), "v"(src) : "memory");
    } else {
        int g = g0 + (lane - 16); if (g > NM - 1) g = NM - 1;
        const void* src = (const void*)(gt_labels + ((size_t)b * NM + g));
        uint32_t lo = (uint32_t)(uintptr_t)(&sLab[lane - 16]);
        asm volatile("global_load_async_to_lds_b32 %0, %1, off"
                     :: "v"(lo), "v"(src) : "memory");
    }
    asm volatile("s_wait_asynccnt 0" ::: "memory");
    __syncthreads();

    const int myLabel = sLab[mrow];            // A-matrix row M = lane%16
    const int a       = a0 + mrow;             // B/C/D column N = lane%16
    const float* bp   = pred_scores + ((size_t)b * NA + a) * NC;

    v8f acc = {};
    #pragma unroll
    for (int k0 = 0; k0 < NC; k0 += 4) {
        const int ka = k0 + (hi ? 2 : 0);
        v2f afrag, bfrag;
        afrag.x = (myLabel == ka)     ? 1.0f : 0.0f;  // one-hot A: 16x4
        afrag.y = (myLabel == ka + 1) ? 1.0f : 0.0f;
        bfrag.x = bp[ka];                              // B: 4x16 = scores^T
        bfrag.y = bp[ka + 1];
        acc = __builtin_amdgcn_wmma_f32_16x16x4_f32(
            false, afrag, false, bfrag, (short)0, acc, false, false);
    }

    // Per-lane epilogue: 8 GT rows for this anchor column.
    const float4 pb = *(const float4*)(pred_bboxes + ((size_t)b * NA + a) * 4);
    const float2 ap = *(const float2*)(anchor_points + (size_t)a * 2);
    const float areaP = (pb.z - pb.x) * (pb.w - pb.y);

    #pragma unroll
    for (int r = 0; r < 8; ++r) {
        const int gl = r + (hi ? 8 : 0);       // D layout: VGPR r -> M=r(+8)
        const int g  = g0 + gl;
        if (g >= NM) continue;
        const float gx1 = sGt[gl * 4 + 0], gy1 = sGt[gl * 4 + 1];
        const float gx2 = sGt[gl * 4 + 2], gy2 = sGt[gl * 4 + 3];
        // IoU
        float lx = fmaxf(gx1, pb.x), ly = fmaxf(gy1, pb.y);
        float rx = fminf(gx2, pb.z), ry = fminf(gy2, pb.w);
        float iw = fmaxf(rx - lx, 0.0f), ih = fmaxf(ry - ly, 0.0f);
        float inter = iw * ih;
        float areaG = (gx2 - gx1) * (gy2 - gy1);
        float iou = inter / (areaG + areaP - inter + EPSF);
        // in_gts: anchor center strictly inside gt box
        float dmin = fminf(fminf(ap.x - gx1, ap.y - gy1),
                           fminf(gx2 - ap.x, gy2 - ap.y));
        float ingts = (dmin > EPSF) ? 1.0f : 0.0f;
        float score = acc[r];
        float i2 = iou * iou;
        float metric = score * (i2 * i2 * i2) * ingts;   // score^1 * iou^6 * in_gts
        size_t off = ((size_t)b * NM + g) * NA + a;
        overlaps[off] = iou;
        metrics[off]  = metric;
    }
}

// ---------------------------------------------------------------- top-k + scatter
// One block (128 threads) per (batch, gt) row: coalesced scan of metrics row,
// register top-13 per thread, LDS selection of global top-13 (stable ties by
// smaller anchor index, matching jax.lax.top_k), then scatter mask_pos.
__device__ inline bool better(float v1, int a1, float v2, int a2) {
    return (v1 > v2) || (v1 == v2 && a1 < a2);
}

__global__ void __launch_bounds__(128)
k_topk(const float* __restrict__ metrics,
       const float* __restrict__ anchor_points,
       const float* __restrict__ gt_bboxes,
       const float* __restrict__ mask_gt,
       int* __restrict__ pos_sum,
       int* __restrict__ agt) {
    const int g = blockIdx.x;
    const int b = blockIdx.y;
    if (mask_gt[b * NM + g] <= 0.0f) return;   // masked row contributes nothing

    const int t = threadIdx.x;
    const float* mrow = metrics + ((size_t)b * NM + g) * NA;

    float tv[TOPK];
    int   ti[TOPK];
    #pragma unroll
    for (int j = 0; j < TOPK; ++j) { tv[j] = -1.0f; ti[j] = 0x7fffffff; }

    for (int a = t; a < NA; a += 128) {
        float v = mrow[a];
        if (better(v, a, tv[TOPK - 1], ti[TOPK - 1])) {
            tv[TOPK - 1] = v; ti[TOPK - 1] = a;
            #pragma unroll
            for (int j = TOPK - 1; j > 0; --j) {
                if (better(tv[j], ti[j], tv[j - 1], ti[j - 1])) {
                    float fv = tv[j]; tv[j] = tv[j - 1]; tv[j - 1] = fv;
                    int   fi = ti[j]; ti[j] = ti[j - 1]; ti[j - 1] = fi;
                } else break;
            }
        }
    }

    __shared__ float sv[128 * TOPK];
    __shared__ int   si[128 * TOPK];
    __shared__ float rv[128];
    __shared__ int   ra[128];
    __shared__ int   rs[128];
    __shared__ int   win[TOPK];

    #pragma unroll
    for (int j = 0; j < TOPK; ++j) { sv[t * TOPK + j] = tv[j]; si[t * TOPK + j] = ti[j]; }
    __syncthreads();

    for (int k = 0; k < TOPK; ++k) {
        float bv = -3.0f; int ba = 0x7fffffff; int bslot = t * TOPK;
        #pragma unroll
        for (int j = 0; j < TOPK; ++j) {
            int s = t * TOPK + j;
            if (better(sv[s], si[s], bv, ba)) { bv = sv[s]; ba = si[s]; bslot = s; }
        }
        rv[t] = bv; ra[t] = ba; rs[t] = bslot;
        __syncthreads();
        for (int off = 64; off > 0; off >>= 1) {
            if (t < off) {
                if (better(rv[t + off], ra[t + off], rv[t], ra[t])) {
                    rv[t] = rv[t + off]; ra[t] = ra[t + off]; rs[t] = rs[t + off];
                }
            }
            __syncthreads();
        }
        if (t == 0) { win[k] = ra[0]; sv[rs[0]] = -2.0f; si[rs[0]] = 0x7fffffff; }
        __syncthreads();
    }

    if (t < TOPK) {
        const int a = win[t];
        const float2 ap = *(const float2*)(anchor_points + (size_t)a * 2);
        const float4 gb = *(const float4*)(gt_bboxes + ((size_t)b * NM + g) * 4);
        float dmin = fminf(fminf(ap.x - gb.x, ap.y - gb.y),
                           fminf(gb.z - ap.x, gb.w - ap.y));
        if (dmin > EPSF) {                      // mask_pos = in_topk * in_gts
            atomicAdd(&pos_sum[b * NA + a], 1);
            atomicMin(&agt[b * NA + a], g);     // argmax of 0/1 row = first set gt
        }
    }
}

// ---------------------------------------------------------------- resolve
__global__ void k_resolve(const float* __restrict__ pred_scores,
                          const int*   __restrict__ gt_labels,
                          const float* __restrict__ overlaps,
                          const int*   __restrict__ pos_sum,
                          int*      __restrict__ agt,
                          float*    __restrict__ aalign,
                          unsigned* __restrict__ pos_align,
                          unsigned* __restrict__ pos_ovl) {
    int i = blockIdx.x * blockDim.x + threadIdx.x;
    if (i >= BS * NA) return;
    const int b = i / NA;
    const int a = i - b * NA;

    const int s = pos_sum[i];
    if (s <= 0) { agt[i] = -1; aalign[i] = 0.0f; return; }

    int g;
    if (s == 1) {
        g = agt[i];
    } else {
        // multi-assigned: resolve to argmax IoU over gts (first occurrence wins)
        float best = -1.0f; int bg = 0;
        for (int gg = 0; gg < NM; ++gg) {
            float o = overlaps[((size_t)b * NM + gg) * NA + a];
            if (o > best) { best = o; bg = gg; }
        }
        g = bg;
    }
    agt[i] = g;

    const float ovl = overlaps[((size_t)b * NM + g) * NA + a];
    const int lab = gt_labels[b * NM + g];
    const float sc = pred_scores[((size_t)b * NA + a) * NC + lab];
    const float o2 = ovl * ovl;
    const float align = sc * (o2 * o2 * o2);     // align_metric at assigned slot
    aalign[i] = align;
    // All values >= 0 so uint bit-pattern max == float max (order-independent).
    atomicMax(&pos_align[b * NM + g], __float_as_uint(align));
    atomicMax(&pos_ovl[b * NM + g], __float_as_uint(ovl));
}

// ---------------------------------------------------------------- per-anchor outputs
__global__ void k_out(const int*   __restrict__ gt_labels,
                      const float* __restrict__ gt_bboxes,
                      const int*   __restrict__ agt,
                      const float* __restrict__ aalign,
                      const unsigned* __restrict__ pos_align,
                      const unsigned* __restrict__ pos_ovl,
                      float* __restrict__ out_labels,
                      float* __restrict__ out_bboxes,
                      float* __restrict__ out_pos,
                      float* __restrict__ normArr,
                      int*   __restrict__ labArr) {
    int i = blockIdx.x * blockDim.x + threadIdx.x;
    if (i >= BS * NA) return;
    const int b = i / NA;

    const int g = agt[i];
    const bool pos = (g >= 0);
    const int gg = pos ? g : 0;                 // argmax over all-zero col -> gt 0
    const int lab = gt_labels[b * NM + gg];

    out_labels[i] = pos ? (float)lab : (float)NC;
    const float4 gb = *(const float4*)(gt_bboxes + ((size_t)b * NM + gg) * 4);
    out_bboxes[(size_t)i * 4 + 0] = gb.x;
    out_bboxes[(size_t)i * 4 + 1] = gb.y;
    out_bboxes[(size_t)i * 4 + 2] = gb.z;
    out_bboxes[(size_t)i * 4 + 3] = gb.w;
    out_pos[i] = pos ? 1.0f : 0.0f;

    float norm = 0.0f;
    if (pos) {
        const float pa = __uint_as_float(pos_align[b * NM + g]);
        const float po = __uint_as_float(pos_ovl[b * NM + g]);
        norm = aalign[i] * po / (pa + EPSF);
    }
    normArr[i] = norm;
    labArr[i] = pos ? lab : -1;
}

// ---------------------------------------------------------------- one-hot * norm scores
__global__ void k_fill(const float* __restrict__ normArr,
                       const int*   __restrict__ labArr,
                       float* __restrict__ out_scores) {
    size_t i = (size_t)blockIdx.x * blockDim.x + threadIdx.x;
    if (i >= (size_t)BS * NA * NC) return;
    const size_t ai = i / NC;
    const int c = (int)(i - ai * NC);
    out_scores[i] = (c == labArr[ai]) ? normArr[ai] : 0.0f;
}

// ---------------------------------------------------------------- launch
extern "C" void kernel_launch(void* const* d_in, const int* in_sizes, int n_in,
                              void* d_out, int out_size, void* d_ws, size_t ws_size,
                              hipStream_t stream) {
    const float* pred_scores   = (const float*)d_in[0];
    const float* pred_bboxes   = (const float*)d_in[1];
    const float* anchor_points = (const float*)d_in[2];
    const int*   gt_labels     = (const int*)d_in[3];
    const float* gt_bboxes     = (const float*)d_in[4];
    const float* mask_gt       = (const float*)d_in[5];

    // Workspace carve-up (~220 MB)
    char* p = (char*)d_ws;
    auto take = [&](size_t bytes) { char* r = p; p += (bytes + 255) & ~size_t(255); return r; };
    float*    overlaps  = (float*)take((size_t)BS * NM * NA * 4);
    float*    metrics   = (float*)take((size_t)BS * NM * NA * 4);
    int*      pos_sum   = (int*)take((size_t)BS * NA * 4);
    int*      agt       = (int*)take((size_t)BS * NA * 4);
    float*    aalign    = (float*)take((size_t)BS * NA * 4);
    float*    normArr   = (float*)take((size_t)BS * NA * 4);
    int*      labArr    = (int*)take((size_t)BS * NA * 4);
    unsigned* pos_align = (unsigned*)take((size_t)BS * NM * 4);
    unsigned* pos_ovl   = (unsigned*)take((size_t)BS * NM * 4);

    // Output layout: labels | bboxes | scores | pos (all fp32, flat, in order)
    float* out_labels = (float*)d_out;
    float* out_bboxes = out_labels + (size_t)BS * NA;
    float* out_scores = out_bboxes + (size_t)BS * NA * 4;
    float* out_pos    = out_scores + (size_t)BS * NA * NC;

    const int nAnch = BS * NA;

    k_init<<<(nAnch + 255) / 256, 256, 0, stream>>>(pos_sum, agt, pos_align, pos_ovl);

    dim3 tgrid(NA / 16, (NM + 15) / 16, BS);
    k_tile<<<tgrid, 32, 0, stream>>>(pred_scores, pred_bboxes, anchor_points,
                                     gt_labels, gt_bboxes, overlaps, metrics);

    dim3 kgrid(NM, BS);
    k_topk<<<kgrid, 128, 0, stream>>>(metrics, anchor_points, gt_bboxes, mask_gt,
                                      pos_sum, agt);

    k_resolve<<<(nAnch + 255) / 256, 256, 0, stream>>>(pred_scores, gt_labels, overlaps,
                                                       pos_sum, agt, aalign,
                                                       pos_align, pos_ovl);

    k_out<<<(nAnch + 255) / 256, 256, 0, stream>>>(gt_labels, gt_bboxes, agt, aalign,
                                                   pos_align, pos_ovl,
                                                   out_labels, out_bboxes, out_pos,
                                                   normArr, labArr);

    size_t nScore = (size_t)BS * NA * NC;
    k_fill<<<(unsigned)((nScore + 255) / 256), 256, 0, stream>>>(normArr, labArr, out_scores);
}